// NeuralODEModel_10230612099786
// MI455X (gfx1250) — compile-verified
//
#include <hip/hip_runtime.h>
#include <stdint.h>

#define B_TOT 8192
#define S_LEN 256
#define HID   64
#define NSUB  4
#define EPS_T 1e-6f
#define SPLIT 4              // lanes cooperating on one batch element
#define KPL   (HID / SPLIT)  // 16 hidden units per lane
#define BLOCK 64             // 2 waves per workgroup -> 512 workgroups

typedef unsigned int u32x4 __attribute__((ext_vector_type(4)));
typedef int          i32x8 __attribute__((ext_vector_type(8)));
typedef int          i32x4 __attribute__((ext_vector_type(4)));

__device__ __forceinline__ float fast_tanh(float x) {
#if __has_builtin(__builtin_amdgcn_tanhf)
    return __builtin_amdgcn_tanhf(x);        // v_tanh_f32 (gfx1250 trans op)
#else
    float r;
    asm("v_tanh_f32 %0, %1" : "=v"(r) : "v"(x));
    return r;
#endif
}

// Cross-lane xor-shuffle inside 4-lane groups, on the VALU pipe (v_permlane16_b32)
// instead of the DS pipe (ds_swizzle), removing DScnt waits from the serial path.
__device__ __forceinline__ float xor1_lane(float v) {
#if __has_builtin(__builtin_amdgcn_permlane16)
    unsigned iv = __float_as_uint(v);
    // sel[i] = i ^ 1 within each 16-lane row
    unsigned r = __builtin_amdgcn_permlane16(iv, iv, 0x67452301u, 0xEFCDAB89u,
                                             false, false);
    return __uint_as_float(r);
#else
    return __shfl_xor(v, 1, 32);
#endif
}
__device__ __forceinline__ float xor2_lane(float v) {
#if __has_builtin(__builtin_amdgcn_permlane16)
    unsigned iv = __float_as_uint(v);
    // sel[i] = i ^ 2 within each 16-lane row
    unsigned r = __builtin_amdgcn_permlane16(iv, iv, 0x54761032u, 0xDCFE98BAu,
                                             false, false);
    return __uint_as_float(r);
#else
    return __shfl_xor(v, 2, 32);
#endif
}

// TDM 1-D copy: nelem f32 from global -> LDS offset (bytes). D# per CDNA5 ISA §8.3-8.5.
__device__ __forceinline__ void tdm_load_1d(uint32_t lds_off_bytes,
                                            const void* gptr, uint32_t nelem) {
#if __has_builtin(__builtin_amdgcn_tensor_load_to_lds)
    uint64_t ga = (uint64_t)(uintptr_t)gptr;
    u32x4 g0;
    g0[0] = 1u;                               // count=1, user mode, gather off
    g0[1] = lds_off_bytes;                    // lds_addr
    g0[2] = (uint32_t)(ga & 0xFFFFFFFFull);   // global_addr[31:0]
    g0[3] = (uint32_t)((ga >> 32) & 0x01FFFFFFu) | (2u << 30); // addr[56:32] | type=2
    i32x8 g1;
    g1[0] = (int)(2u << 16);                  // data_size = 4 bytes; no barrier/iterate/pad
    g1[1] = (int)((nelem & 0xFFFFu) << 16);   // tensor_dim0[15:0]
    g1[2] = (int)(((nelem >> 16) & 0xFFFFu) | (1u << 16)); // dim0 hi | tensor_dim1 = 1
    g1[3] = (int)(nelem << 16);               // dim1 hi = 0 | tile_dim0 = nelem
    g1[4] = 1;                                // tile_dim1 = 1, tile_dim2 = 0
    g1[5] = (int)nelem;                       // tensor_dim0_stride
    g1[6] = 0;
    g1[7] = 0;
    i32x4 z4 = {0, 0, 0, 0};                  // D# group 2 (unused, <=2D tile)
    i32x8 z8 = {0, 0, 0, 0, 0, 0, 0, 0};      // extra group (6-arg clang-23 form), zero-filled
    __builtin_amdgcn_tensor_load_to_lds(g0, g1, z4, z4, z8, 0);
#else
    (void)lds_off_bytes; (void)gptr; (void)nelem; // plain prefill already staged data
#endif
}

__device__ __forceinline__ void wait_tensor0() {
#if __has_builtin(__builtin_amdgcn_s_wait_tensorcnt)
    __builtin_amdgcn_s_wait_tensorcnt(0);
#else
    asm volatile("s_wait_tensorcnt 0x0" ::: "memory");
#endif
}

__global__ __launch_bounds__(BLOCK)
void node_rk4_kernel(const float* __restrict__ x,  const float* __restrict__ W1,
                     const float* __restrict__ b1, const float* __restrict__ W2,
                     const float* __restrict__ b2, const float* __restrict__ fcw,
                     const float* __restrict__ fcb, float* __restrict__ out) {
    // smem layout (floats): [0..127] W1 (2x64 row-major), [128..191] b1, [192..319] W2 (64x2)
    __shared__ float smem[320];

    // 1) cooperative plain prefill (correctness backstop)
    for (int i = threadIdx.x; i < 128; i += BLOCK) smem[i]       = W1[i];
    for (int i = threadIdx.x; i < 64;  i += BLOCK) smem[128 + i] = b1[i];
    for (int i = threadIdx.x; i < 128; i += BLOCK) smem[192 + i] = W2[i];
    __syncthreads();

    // 2) TDM rewrite of the same bytes (async tensor path; wave 0 only)
    if (threadIdx.x < 32) {
        uint32_t base = (uint32_t)(uintptr_t)&smem[0];   // LDS flat addr low 32b = LDS offset
        tdm_load_1d(base + 0u,   W1, 128u);
        tdm_load_1d(base + 512u, b1,  64u);
        tdm_load_1d(base + 768u, W2, 128u);
        wait_tensor0();
    }
    __syncthreads();

    const int tid = blockIdx.x * BLOCK + threadIdx.x;
    const int e   = tid >> 2;     // batch element
    const int sub = tid & 3;      // which 16-wide hidden slice this lane owns

    // hoist this lane's weight slice into VGPRs (loop-invariant, 80 regs)
    float wa[KPL], wb[KPL], wc[KPL], wp[KPL], wq[KPL];
    const int k0 = sub * KPL;
#pragma unroll
    for (int j = 0; j < KPL; ++j) {
        const int k = k0 + j;
        wa[j] = smem[k];            // W1[0][k]
        wb[j] = smem[64 + k];       // W1[1][k]
        wc[j] = smem[128 + k];      // b1[k]
        wp[j] = smem[192 + 2 * k];      // W2[k][0]
        wq[j] = smem[192 + 2 * k + 1];  // W2[k][1]
    }
    const float b20 = b2[0], b21 = b2[1];
    const float fw0 = fcw[0], fw1 = fcw[1], fb = fcb[0];

    // f(y) = tanh(y@W1 + b1) @ W2 + b2 ; partial over 16 hidden, 4-lane all-reduce
    auto feval = [&](float u, float v, float& zx, float& zy) {
        float z0a = 0.f, z0b = 0.f, z1a = 0.f, z1b = 0.f;
#pragma unroll
        for (int j = 0; j < KPL; j += 2) {
            float g0 = fmaf(v, wb[j],     fmaf(u, wa[j],     wc[j]));
            float g1 = fmaf(v, wb[j + 1], fmaf(u, wa[j + 1], wc[j + 1]));
            float t0 = fast_tanh(g0);
            float t1 = fast_tanh(g1);
            z0a = fmaf(t0, wp[j],     z0a);
            z1a = fmaf(t0, wq[j],     z1a);
            z0b = fmaf(t1, wp[j + 1], z0b);
            z1b = fmaf(t1, wq[j + 1], z1b);
        }
        float z0 = z0a + z0b;
        float z1 = z1a + z1b;
        z0 += xor1_lane(z0);           // all-reduce within the 4-lane group (VALU pipe)
        z0 += xor2_lane(z0);
        z1 += xor1_lane(z1);
        z1 += xor2_lane(z1);
        zx = z0 + b20;
        zy = z1 + b21;
    };

    const float* xe = x + (size_t)e * (S_LEN * 3);
    float y0 = xe[0];
    float y1 = xe[1];
    float tp = xe[2];               // t_fixed carry

    for (int s = 1; s < S_LEN; ++s) {
        const float tj = xe[s * 3 + 2];
        {   // hide the next intervals' dependent t-loads behind this interval's compute
            const int sp = (s + 8 < S_LEN) ? (s + 8) : (S_LEN - 1);
            __builtin_prefetch(xe + sp * 3 + 2, 0, 3);   // global_prefetch_b8
        }
        const float tn = (tj <= tp) ? (tp + EPS_T) : tj;
        const float h  = (tn - tp) * (1.0f / (float)NSUB);
        tp = tn;
        const float hh = 0.5f * h;
        const float w6 = h * (1.0f / 6.0f);
#pragma unroll 1
        for (int ss = 0; ss < NSUB; ++ss) {
            float k1x, k1y, k2x, k2y, k3x, k3y, k4x, k4y;
            feval(y0,                y1,                k1x, k1y);
            feval(fmaf(hh, k1x, y0), fmaf(hh, k1y, y1), k2x, k2y);
            feval(fmaf(hh, k2x, y0), fmaf(hh, k2y, y1), k3x, k3y);
            feval(fmaf(h,  k3x, y0), fmaf(h,  k3y, y1), k4x, k4y);
            y0 = fmaf(w6, k1x + 2.0f * (k2x + k3x) + k4x, y0);
            y1 = fmaf(w6, k1y + 2.0f * (k2y + k3y) + k4y, y1);
        }
    }

    if (sub == 0) {
        out[e] = fmaf(y0, fw0, fmaf(y1, fw1, fb));   // yT @ fc_w.T + fc_b
    }
}

extern "C" void kernel_launch(void* const* d_in, const int* in_sizes, int n_in,
                              void* d_out, int out_size, void* d_ws, size_t ws_size,
                              hipStream_t stream) {
    (void)in_sizes; (void)n_in; (void)out_size; (void)d_ws; (void)ws_size;
    const float* x   = (const float*)d_in[0];
    const float* W1  = (const float*)d_in[1];
    const float* b1  = (const float*)d_in[2];
    const float* W2  = (const float*)d_in[3];
    const float* b2  = (const float*)d_in[4];
    const float* fcw = (const float*)d_in[5];
    const float* fcb = (const float*)d_in[6];
    dim3 grid((B_TOT * SPLIT) / BLOCK);
    dim3 block(BLOCK);
    node_rk4_kernel<<<grid, block, 0, stream>>>(x, W1, b1, W2, b2, fcw, fcb,
                                                (float*)d_out);
}